// DepthwiseStencil3D_25821343383820
// MI455X (gfx1250) — compile-verified
//
#include <hip/hip_runtime.h>
#include <hip/hip_bf16.h>

// DepthwiseStencil3D: output = six zero-padded shifted copies of the input.
// Pure data movement: 128 MiB read + 768 MiB written. Strategy for MI455X:
//  - input (128 MiB) fits in the 192 MB L2 -> keep it resident by streaming
//    the write-only output with non-temporal (TH=NT) b128 stores
//  - one thread per 4 contiguous w positions, computes all 6 taps:
//    5x global_load_b128 + 2 scalar edge loads -> 6x global_store_b128 nt
//  - global_prefetch_b8 warms L2 for the d+2 plane

typedef float v4f __attribute__((ext_vector_type(4)));

#define NC   16
#define ND   128
#define NH   128
#define NW   128
#define HW   (NH * NW)          // 16384
#define DHW  (ND * NH * NW)     // 2097152

__global__ __launch_bounds__(256) void
DepthwiseStencil3D_kernel(const float* __restrict__ x, float* __restrict__ out) {
    const int t  = blockIdx.x * 256 + threadIdx.x;   // [0, NC*ND*NH*NW/4)
    const int w4 = t & 31;                           // NW/4 = 32 float4 per row
    const int h  = (t >> 5)  & 127;
    const int d  = (t >> 12) & 127;
    const int c  =  t >> 19;                         // 0..15
    const int w0 = w4 << 2;

    const float* p = x + (size_t)c * DHW + (size_t)d * HW + h * NW + w0;

    // Warm L2 for the d+2 plane (this block's d-neighbors of upcoming work).
    if (d + 2 < ND) {
        __builtin_prefetch(p + 2 * HW, 0, 3);
    }

    const v4f zero = (v4f)0.0f;
    v4f   v  = *(const v4f*)p;                               // x[w0..w0+3]
    float xl = (w0 > 0)       ? p[-1] : 0.0f;                // x[w0-1]
    float xr = (w0 + 4 < NW)  ? p[4]  : 0.0f;                // x[w0+4]
    v4f   hp = (h < NH - 1)   ? *(const v4f*)(p + NW) : zero; // h+1 row
    v4f   hm = (h > 0)        ? *(const v4f*)(p - NW) : zero; // h-1 row
    v4f   dp = (d < ND - 1)   ? *(const v4f*)(p + HW) : zero; // d+1 plane
    v4f   dm = (d > 0)        ? *(const v4f*)(p - HW) : zero; // d-1 plane

    v4f k0 = {v.y, v.z, v.w, xr};   // tap w+1: y[w] = x[w+1]
    v4f k1 = {xl, v.x, v.y, v.z};   // tap w-1: y[w] = x[w-1]

    // Flat output channel n = c*6 + k holds shift k of input channel c
    // (the reference's .view(B,6,C,...) is a reinterpretation of that layout).
    float* q = out + ((size_t)c * 6) * DHW + (size_t)d * HW + h * NW + w0;
    __builtin_nontemporal_store(k0, (v4f*)(q + 0 * DHW));
    __builtin_nontemporal_store(k1, (v4f*)(q + 1 * DHW));
    __builtin_nontemporal_store(hp, (v4f*)(q + 2 * DHW));
    __builtin_nontemporal_store(hm, (v4f*)(q + 3 * DHW));
    __builtin_nontemporal_store(dp, (v4f*)(q + 4 * DHW));
    __builtin_nontemporal_store(dm, (v4f*)(q + 5 * DHW));
}

extern "C" void kernel_launch(void* const* d_in, const int* in_sizes, int n_in,
                              void* d_out, int out_size, void* d_ws, size_t ws_size,
                              hipStream_t stream) {
    const float* x   = (const float*)d_in[0];
    float*       out = (float*)d_out;

    // One thread per float4 of one channel's spatial volume, all 6 taps each.
    const int total_threads = NC * ND * NH * (NW / 4);   // 8,388,608
    const int block = 256;                               // 8 wave32s
    const int grid  = total_threads / block;             // 32768

    DepthwiseStencil3D_kernel<<<grid, block, 0, stream>>>(x, out);
}